// CausalDepthwiseConv1d_57475252355405
// MI455X (gfx1250) — compile-verified
//
#include <hip/hip_runtime.h>

// Causal depthwise conv1d, B=8, T=4096, C=2048, K=4, fp32.
// Memory-bound (AI ~= 1 FLOP/byte): design = stream x once with b128 NT
// loads, sliding window in registers along T, b128 NT stores. Weights/bias
// staged via CDNA5 async-to-LDS (ASYNCcnt) path using inline asm.

#define BDIM   256
#define VPT    4       // channels per thread (float4)
#define T_TILE 64      // time steps per block
#define BB     8
#define TT     4096
#define CC     2048
#define KK     4

typedef __attribute__((ext_vector_type(4))) float v4f;

#if __has_builtin(__builtin_amdgcn_s_wait_asynccnt)
#define HAVE_WAIT_ASYNC 1
#endif

// CDNA5 §15.18 op 98 GLOBAL_LOAD_ASYNC_TO_LDS_B128, GV mode:
//   each lane copies 16 bytes from its global address to its LDS address
// VDST holds the per-lane LDS byte address; VADDR holds the 64-bit global
// address. For generic pointers in the LDS aperture, addr[31:0] IS the
// wave-relative LDS byte offset (§10.2), so truncation gives us VDST.
__device__ __forceinline__ void async_copy_b128(const void* gsrc, void* lds_dst) {
  unsigned lds_off = (unsigned)(unsigned long long)lds_dst;  // low 32 bits = LDS offset
  asm volatile("global_load_async_to_lds_b128 %0, %1, off"
               :: "v"(lds_off), "v"((unsigned long long)gsrc)
               : "memory");
}

__device__ __forceinline__ void wait_async0() {
#if defined(HAVE_WAIT_ASYNC)
  __builtin_amdgcn_s_wait_asynccnt(0);
#else
  asm volatile("s_wait_asynccnt 0" ::: "memory");
#endif
}

__device__ __forceinline__ v4f nt_load4(const float* p) {
  return __builtin_nontemporal_load((const v4f*)p);
}
__device__ __forceinline__ void nt_store4(float* p, v4f v) {
  __builtin_nontemporal_store(v, (v4f*)p);
}

__global__ __launch_bounds__(BDIM)
void CausalDepthwiseConv1d_kernel(const float* __restrict__ x,
                                  const float* __restrict__ w,     // [C,1,K]
                                  const float* __restrict__ bias,  // [C]
                                  float* __restrict__ out) {
  __shared__ v4f wlds[BDIM * VPT];  // per channel: taps k=0..3
  __shared__ v4f blds[BDIM];        // 4 bias values per thread

  const int tid = threadIdx.x;
  const int c4  = blockIdx.y * (BDIM * VPT) + tid * VPT;  // first of 4 channels
  const int b   = blockIdx.z;
  const int t0  = blockIdx.x * T_TILE;

  // ---- Stage weights + bias into LDS via async-to-LDS (ASYNCcnt path) ----
#pragma unroll
  for (int j = 0; j < VPT; ++j)
    async_copy_b128(w + (size_t)(c4 + j) * KK, &wlds[tid * VPT + j]);
  async_copy_b128(bias + c4, &blds[tid]);
  wait_async0();

  // Transpose 4 channel-major tap vectors into 4 tap-major channel vectors.
  const v4f a0 = wlds[tid * VPT + 0];
  const v4f a1 = wlds[tid * VPT + 1];
  const v4f a2 = wlds[tid * VPT + 2];
  const v4f a3 = wlds[tid * VPT + 3];
  const v4f bi = blds[tid];
  const v4f wk0 = {a0.x, a1.x, a2.x, a3.x};  // multiplies x[t-3]
  const v4f wk1 = {a0.y, a1.y, a2.y, a3.y};  // x[t-2]
  const v4f wk2 = {a0.z, a1.z, a2.z, a3.z};  // x[t-1]
  const v4f wk3 = {a0.w, a1.w, a2.w, a3.w};  // x[t]

  // ---- Stream the time strip with a register sliding window ----
  const size_t base = ((size_t)b * TT + t0) * CC + c4;
  const float* xp = x + base;
  float*       op = out + base;

  v4f h0, h1, h2;  // x[t-3], x[t-2], x[t-1]
  if (t0 == 0) {
    h0 = (v4f){0.f, 0.f, 0.f, 0.f};
    h1 = h0;
    h2 = h0;
  } else {
    h0 = nt_load4(xp - 3 * CC);
    h1 = nt_load4(xp - 2 * CC);
    h2 = nt_load4(xp - 1 * CC);
  }

#pragma unroll 4
  for (int i = 0; i < T_TILE; ++i) {
    v4f h3 = nt_load4(xp + (size_t)i * CC);
    v4f o  = bi + wk0 * h0 + wk1 * h1 + wk2 * h2 + wk3 * h3;
    nt_store4(op + (size_t)i * CC, o);
    h0 = h1; h1 = h2; h2 = h3;
  }
}

extern "C" void kernel_launch(void* const* d_in, const int* in_sizes, int n_in,
                              void* d_out, int out_size, void* d_ws, size_t ws_size,
                              hipStream_t stream) {
  const float* x    = (const float*)d_in[0];
  const float* w    = (const float*)d_in[1];
  const float* bias = (const float*)d_in[2];
  float*       out  = (float*)d_out;
  (void)in_sizes; (void)n_in; (void)out_size; (void)d_ws; (void)ws_size;

  dim3 grid(TT / T_TILE, CC / (BDIM * VPT), BB);  // (64, 2, 8)
  CausalDepthwiseConv1d_kernel<<<grid, dim3(BDIM), 0, stream>>>(x, w, bias, out);
}